// HedgerDeploymentPPO_31035433681220
// MI455X (gfx1250) — compile-verified
//
#include <hip/hip_runtime.h>
#include <hip/hip_bf16.h>
#include <math.h>

typedef __attribute__((ext_vector_type(2))) float v2f;
typedef __attribute__((ext_vector_type(8))) float v8f;

#define EPS_F 1.1920929e-07f

// ---------------------------------------------------------------------------
// fp32 WMMA GEMM: C[M,N] = act( scale * (A[M,K] @ op(B)) )
//   TRANSB==0: B is [K,N] row-major.  TRANSB==1: B is [N,K] row-major (B^T).
// One 16x64 output tile per wave32: 4 independent v8f accumulators so the
// 4 V_WMMA_F32_16X16X4_F32 per K-step pipeline without D->D RAW stalls, and
// the A fragment is reused 4x per load.  Compile-time flags keep the K-loop
// branch-free (previous round showed runtime flags scalarized the loop).
// ---------------------------------------------------------------------------
template <int TRANSB, int SIGMOID>
__global__ void gemm_f32_wmma_kernel(const float* __restrict__ A,
                                     const float* __restrict__ B,
                                     float* __restrict__ C,
                                     int Mn, int Nn, int Kn, float scale)
{
    const int lane        = threadIdx.x & 31;
    const int waveInBlock = threadIdx.x >> 5;
    const int wavesPerBlk = blockDim.x >> 5;
    const int gwave       = blockIdx.x * wavesPerBlk + waveInBlock;
    const int tilesN64    = Nn >> 6;               // N tiled by 64
    const int tilesM      = Mn >> 4;
    if (gwave >= tilesM * tilesN64) return;        // wave-uniform: EXEC stays full
    const int tm = gwave / tilesN64;
    const int tn = gwave % tilesN64;

    // ISA 16x4 fp32 A layout: lanes 0-15 -> K={0,1}, lanes 16-31 -> K={2,3}
    const int half = lane >> 4;
    const int lid  = lane & 15;
    const int arow = tm * 16 + lid;                // A: M striped over lanes

    const float* __restrict__ arowp = A + (size_t)arow * Kn;

    // Per-subtile B base pointers (lane-dependent part folded in once).
    const float* __restrict__ bp[4];
    #pragma unroll
    for (int t = 0; t < 4; ++t) {
        const int bcol = tn * 64 + t * 16 + lid;
        bp[t] = TRANSB ? (B + (size_t)bcol * Kn)   // walk along a row of B^T
                       : (B + bcol);               // walk down a column of B
    }

    v8f acc[4];
    #pragma unroll
    for (int t = 0; t < 4; ++t) acc[t] = (v8f){0.f,0.f,0.f,0.f,0.f,0.f,0.f,0.f};

    #pragma unroll 2
    for (int k = 0; k < Kn; k += 4) {
        const int k0 = k + half * 2;
        v2f a;
        a.x = arowp[k0];
        a.y = arowp[k0 + 1];
        #pragma unroll
        for (int t = 0; t < 4; ++t) {
            v2f b;
            if (TRANSB) {
                b.x = bp[t][k0];
                b.y = bp[t][k0 + 1];
            } else {
                b.x = bp[t][(size_t)(k0)     * Nn];
                b.y = bp[t][(size_t)(k0 + 1) * Nn];
            }
            // 8 args: (neg_a, A, neg_b, B, c_mod, C, reuse_a, reuse_b)
            acc[t] = __builtin_amdgcn_wmma_f32_16x16x4_f32(
                         false, a, false, b, (short)0, acc[t], false, false);
        }
    }

    // C/D layout: VGPR r holds row r (lanes 0-15) and row r+8 (lanes 16-31)
    #pragma unroll
    for (int t = 0; t < 4; ++t) {
        const int bcol = tn * 64 + t * 16 + lid;
        #pragma unroll
        for (int r = 0; r < 8; ++r) {
            const int row = tm * 16 + r + half * 8;
            float v = acc[t][r] * scale;
            if (SIGMOID) v = 1.0f / (1.0f + __expf(-v));
            C[(size_t)row * Nn + bcol] = v;
        }
    }
}

// ---------------------------------------------------------------------------
// Per-node residual scan: one thread per physical node p.  The residual
// recurrence is independent of probs, so "allowed[i,p]" (residual BEFORE
// step i >= mem_i, or p==cloud) can be fully precomputed in parallel over p.
// ---------------------------------------------------------------------------
__global__ void residual_allowed_kernel(const float* __restrict__ model_mem,
                                        const float* __restrict__ mem_capacity,
                                        const float* __restrict__ mem_util_seq,
                                        const unsigned char* __restrict__ deploy_mask,
                                        unsigned char* __restrict__ allowed,
                                        int Ms, int Np, int T)
{
    const int p = blockIdx.x * blockDim.x + threadIdx.x;
    if (p >= Np) return;
    const int cloud = Np - 1;
    float residual = mem_capacity[p] * (1.0f - mem_util_seq[(size_t)p * T + (T - 1)]);
    for (int i = 0; i < Ms; ++i) {
        const float  mi  = model_mem[i];
        const size_t idx = (size_t)i * Np + p;
        if (i + 8 < Ms)  // hide the 2048-stride latency -> global_prefetch_b8
            __builtin_prefetch(&deploy_mask[idx + (size_t)8 * Np], 0, 1);
        allowed[idx] = ((p == cloud) || (residual >= mi)) ? 1 : 0;
        if (deploy_mask[idx] && (p != cloud))
            residual = fmaxf(residual - mi, 0.0f);
    }
}

template <int BS>
__device__ inline float block_reduce_sum(float v, float* red)
{
    const int tid = threadIdx.x;
    red[tid] = v;
    __syncthreads();
    for (int s = BS >> 1; s > 0; s >>= 1) {
        if (tid < s) red[tid] += red[tid + s];
        __syncthreads();
    }
    const float r = red[0];
    __syncthreads();
    return r;
}

// One block per row i: rowsum[i] = sum_p (allowed ? probs : 0)
__global__ void rowsum_kernel(const float* __restrict__ probs,
                              const unsigned char* __restrict__ allowed,
                              float* __restrict__ rowsum, int Np)
{
    __shared__ float red[256];
    const int i = blockIdx.x;
    float s = 0.0f;
    for (int p = threadIdx.x; p < Np; p += 256) {
        const size_t idx = (size_t)i * Np + p;
        s += allowed[idx] ? probs[idx] : 0.0f;
    }
    s = block_reduce_sum<256>(s, red);
    if (threadIdx.x == 0) rowsum[i] = s;
}

// One block per row i: logp row-sum and entropy row-mean.
__global__ void loss_kernel(const float* __restrict__ probs,
                            const unsigned char* __restrict__ allowed,
                            const unsigned char* __restrict__ deploy_mask,
                            const float* __restrict__ rowsum,
                            float* __restrict__ lp_row,
                            float* __restrict__ en_row, int Np)
{
    __shared__ float red[256];
    const int  i        = blockIdx.x;
    const int  cloud    = Np - 1;
    const bool fallback = (rowsum[i] == 0.0f);
    float lp = 0.0f, en = 0.0f;
    for (int p = threadIdx.x; p < Np; p += 256) {
        const size_t idx = (size_t)i * Np + p;
        float pm = allowed[idx] ? probs[idx] : 0.0f;
        if (fallback) pm = (p == cloud) ? 1.0f : 0.0f;
        const float pc  = fminf(fmaxf(pm, EPS_F), 1.0f - EPS_F);
        const float lg  = logf(pc);
        const float l1  = log1pf(-pc);
        const float act = deploy_mask[idx] ? 1.0f : 0.0f;
        lp += act * lg + (1.0f - act) * l1;
        en += -(pc * lg + (1.0f - pc) * l1);
    }
    lp = block_reduce_sum<256>(lp, red);
    en = block_reduce_sum<256>(en, red);
    if (threadIdx.x == 0) {
        lp_row[i] = lp;
        en_row[i] = en / (float)Np;   // jnp.mean over nodes
    }
}

// Single block, 512 threads: column means of h_s/h_p, critic value, and
// deterministic reduction of the per-row lp/en partials.
__global__ void finalize_kernel(const float* __restrict__ lp_row,
                                const float* __restrict__ en_row,
                                const float* __restrict__ h_s,
                                const float* __restrict__ h_p,
                                const float* __restrict__ w_v,
                                const float* __restrict__ b_v,
                                float* __restrict__ out,
                                int Ms, int Np, int Dd)
{
    __shared__ float red[512];
    const int d = threadIdx.x;            // 0..511

    float cs = 0.0f, cp = 0.0f;
    for (int i = 0; i < Ms; ++i) cs += h_s[(size_t)i * Dd + d];
    for (int i = 0; i < Np; ++i) cp += h_p[(size_t)i * Dd + d];
    float contrib = (cs / (float)Ms) * w_v[d] + (cp / (float)Np) * w_v[Dd + d];
    float value   = block_reduce_sum<512>(contrib, red);

    float lp = 0.0f, en = 0.0f;
    for (int i = d; i < Ms; i += 512) { lp += lp_row[i]; en += en_row[i]; }
    lp = block_reduce_sum<512>(lp, red);
    en = block_reduce_sum<512>(en, red);

    if (d == 0) {
        out[0] = lp;
        out[1] = en;
        out[2] = value + b_v[0];
    }
}

// ---------------------------------------------------------------------------
extern "C" void kernel_launch(void* const* d_in, const int* in_sizes, int n_in,
                              void* d_out, int out_size, void* d_ws, size_t ws_size,
                              hipStream_t stream)
{
    (void)in_sizes; (void)n_in; (void)out_size; (void)ws_size;

    constexpr int Ms = 4096, Np = 2048, F = 1024, D = 512, T = 16;

    const float*         logic_x      = (const float*)d_in[0];
    const float*         model_mem    = (const float*)d_in[1];
    const float*         phys_x       = (const float*)d_in[2];
    const float*         mem_capacity = (const float*)d_in[3];
    const float*         mem_util_seq = (const float*)d_in[4];
    const unsigned char* deploy_mask  = (const unsigned char*)d_in[5];  // jax bool_ = 1 byte
    const float*         W_s          = (const float*)d_in[6];
    const float*         W_p          = (const float*)d_in[7];
    const float*         W_bil        = (const float*)d_in[8];
    const float*         w_v          = (const float*)d_in[9];
    const float*         b_v          = (const float*)d_in[10];
    float*               out          = (float*)d_out;

    // Workspace layout (bytes).  Total ~60 MB -> resident in 192 MB L2.
    char* ws = (char*)d_ws;
    float*         h_s     = (float*)(ws);                         //  8 MB
    float*         h_p     = (float*)(ws + 8388608);               //  4 MB
    float*         M1      = (float*)(ws + 12582912);              //  8 MB
    float*         probs   = (float*)(ws + 20971520);              // 32 MB
    unsigned char* allowed = (unsigned char*)(ws + 54525952);      //  8 MB
    float*         rowsum  = (float*)(ws + 62914560);              // 16 KB
    float*         lp_row  = (float*)(ws + 62930944);              // 16 KB
    float*         en_row  = (float*)(ws + 62947328);              // 16 KB

    // 1) per-node residual scan -> allowed[i,p]
    residual_allowed_kernel<<<(Np + 255) / 256, 256, 0, stream>>>(
        model_mem, mem_capacity, mem_util_seq, deploy_mask, allowed, Ms, Np, T);

    auto grid_for = [](int M, int N) {
        const int tiles = (M >> 4) * (N >> 6);     // 16x64 tile per wave
        return (tiles + 3) / 4;                    // 4 waves per 128-thread block
    };

    const float inv_sqrt_d = 0.044194173824159216f;  // 1/sqrt(512)

    // 2) h_s = logic_x @ W_s            [4096,1024]x[1024,512]
    gemm_f32_wmma_kernel<0, 0><<<grid_for(Ms, D), 128, 0, stream>>>(
        logic_x, W_s, h_s, Ms, D, F, 1.0f);
    // 3) h_p = phys_x @ W_p             [2048,1024]x[1024,512]
    gemm_f32_wmma_kernel<0, 0><<<grid_for(Np, D), 128, 0, stream>>>(
        phys_x, W_p, h_p, Np, D, F, 1.0f);
    // 4) M1 = h_s @ W_bil               [4096,512]x[512,512]
    gemm_f32_wmma_kernel<0, 0><<<grid_for(Ms, D), 128, 0, stream>>>(
        h_s, W_bil, M1, Ms, D, D, 1.0f);
    // 5) probs = sigmoid((M1 @ h_p^T) / sqrt(D))   [4096,512]x[512,2048]
    gemm_f32_wmma_kernel<1, 1><<<grid_for(Ms, Np), 128, 0, stream>>>(
        M1, h_p, probs, Ms, Np, D, inv_sqrt_d);

    // 6) masked row sums (for the onehot_cloud fallback)
    rowsum_kernel<<<Ms, 256, 0, stream>>>(probs, allowed, rowsum, Np);

    // 7) per-row logp / entropy partials
    loss_kernel<<<Ms, 256, 0, stream>>>(probs, allowed, deploy_mask, rowsum,
                                        lp_row, en_row, Np);

    // 8) critic value + final deterministic reductions -> out[0..2]
    finalize_kernel<<<1, 512, 0, stream>>>(lp_row, en_row, h_s, h_p,
                                           w_v, b_v, out, Ms, Np, D);
}